// HybridVisionBlock_4690104287390
// MI455X (gfx1250) — compile-verified
//
#include <hip/hip_runtime.h>
#include <hip/hip_bf16.h>
#include <math.h>

typedef __attribute__((ext_vector_type(2))) float v2f;
typedef __attribute__((ext_vector_type(8))) float v8f;

#define B_SZ 8
#define G 14
#define LTOK 196
#define D 384
#define NH 12
#define HD 32
#define E 768
#define NS 16
#define DT_RANK 24
#define FFN_DIM 1536
#define MROWS (B_SZ * LTOK)   // 1568

// ---------------------------------------------------------------------------
// Generic fp32 WMMA GEMM:  C[M,N] = A[M,K] @ W[N,K]^T (+bias, +activation)
// BM=64, BN=64, BK=16; 128 threads = 4 waves, wave w owns rows [16w,16w+16)
// epi: 0=none 1=bias 2=bias+gelu(exact) 3=bias+softplus
// ---------------------------------------------------------------------------
#define BM 64
#define BN 64
#define BK 16

__global__ __launch_bounds__(128) void gemm_wmma_f32(
    const float* __restrict__ A, int lda,
    const float* __restrict__ W, int ldb,
    float* __restrict__ C, int ldc,
    const float* __restrict__ bias,
    int M, int N, int K, int epi)
{
  __shared__ float sA[BM * BK];
  __shared__ float sB[BN * BK];
  const int tid  = threadIdx.x;
  const int wave = tid >> 5;
  const int lane = tid & 31;
  const int half = lane >> 4;   // 0: K=k,k+1 ; 1: K=k+2,k+3
  const int lm   = lane & 15;
  const int m0 = blockIdx.y * BM;
  const int n0 = blockIdx.x * BN;

  v8f c0 = {}, c1 = {}, c2 = {}, c3 = {};

  for (int k0 = 0; k0 < K; k0 += BK) {
    for (int i = tid; i < BM * BK; i += 128) {
      int r = i >> 4, cc = i & 15;
      int gm = m0 + r, gk = k0 + cc;
      sA[i] = (gm < M && gk < K) ? A[(size_t)gm * lda + gk] : 0.0f;
    }
    for (int i = tid; i < BN * BK; i += 128) {
      int r = i >> 4, cc = i & 15;
      int gn = n0 + r, gk = k0 + cc;
      sB[i] = (gn < N && gk < K) ? W[(size_t)gn * ldb + gk] : 0.0f;
    }
    // Warm L2/L1 for the next K panel while this chunk computes
    // (lowered to global_prefetch_b8; speculative, OOB-safe, but clamp anyway).
    if (k0 + BK < K) {
      int r = tid & 63;
      int gm = m0 + r; if (gm >= M) gm = M - 1;
      int gn = n0 + r; if (gn >= N) gn = N - 1;
      __builtin_prefetch(&A[(size_t)gm * lda + (k0 + BK)], 0, 1);
      __builtin_prefetch(&W[(size_t)gn * ldb + (k0 + BK)], 0, 1);
    }
    __syncthreads();
#pragma unroll
    for (int kk = 0; kk < BK; kk += 4) {
      const int ks = kk + 2 * half;
      v2f a;
      a.x = sA[(16 * wave + lm) * BK + ks];
      a.y = sA[(16 * wave + lm) * BK + ks + 1];
      v2f b;
      b.x = sB[(0 * 16 + lm) * BK + ks];
      b.y = sB[(0 * 16 + lm) * BK + ks + 1];
      c0 = __builtin_amdgcn_wmma_f32_16x16x4_f32(false, a, false, b, (short)0, c0, false, false);
      b.x = sB[(1 * 16 + lm) * BK + ks];
      b.y = sB[(1 * 16 + lm) * BK + ks + 1];
      c1 = __builtin_amdgcn_wmma_f32_16x16x4_f32(false, a, false, b, (short)0, c1, false, false);
      b.x = sB[(2 * 16 + lm) * BK + ks];
      b.y = sB[(2 * 16 + lm) * BK + ks + 1];
      c2 = __builtin_amdgcn_wmma_f32_16x16x4_f32(false, a, false, b, (short)0, c2, false, false);
      b.x = sB[(3 * 16 + lm) * BK + ks];
      b.y = sB[(3 * 16 + lm) * BK + ks + 1];
      c3 = __builtin_amdgcn_wmma_f32_16x16x4_f32(false, a, false, b, (short)0, c3, false, false);
    }
    __syncthreads();
  }

  // C/D layout: VGPR r, lanes0-15 -> M=r, N=lane ; lanes16-31 -> M=r+8, N=lane-16
  const int mbase = m0 + 16 * wave + 8 * half;
#pragma unroll
  for (int nt = 0; nt < 4; ++nt) {
    v8f cc;
    if (nt == 0) cc = c0; else if (nt == 1) cc = c1; else if (nt == 2) cc = c2; else cc = c3;
    int col = n0 + nt * 16 + lm;
    if (col < N) {
      float bv = (epi != 0 && bias) ? bias[col] : 0.0f;
#pragma unroll
      for (int r = 0; r < 8; ++r) {
        int row = mbase + r;
        if (row < M) {
          float v = cc[r] + bv;
          if (epi == 2) {
            v = 0.5f * v * (1.0f + erff(v * 0.70710678118654752f));
          } else if (epi == 3) {
            v = (v > 20.0f) ? v : log1pf(expf(v));
          }
          C[(size_t)row * ldc + col] = v;
        }
      }
    }
  }
}

// ---------------------------------------------------------------------------
// Router: alpha = sigmoid( gelu([x,entropy] @ W1^T + b1) @ W2^T + b2 )
// ---------------------------------------------------------------------------
__global__ __launch_bounds__(128) void router_kernel(
    const float* __restrict__ x, const float* __restrict__ entropy,
    const float* __restrict__ w1, const float* __restrict__ b1,
    const float* __restrict__ w2, const float* __restrict__ b2,
    float* __restrict__ alpha)
{
  __shared__ float sh[96];
  const int row = blockIdx.x;
  const int tid = threadIdx.x;
  if (tid < 96) {
    const float* wr = w1 + (size_t)tid * (D + 1);
    const float* xr = x + (size_t)row * D;
    float acc = 0.0f;
    for (int k = 0; k < D; ++k) acc += xr[k] * wr[k];
    acc += entropy[row] * wr[D];
    acc += b1[tid];
    float g = 0.5f * acc * (1.0f + erff(acc * 0.70710678118654752f));
    sh[tid] = g * w2[tid];
  }
  __syncthreads();
  if (tid == 0) {
    float s = 0.0f;
    for (int i = 0; i < 96; ++i) s += sh[i];
    s += b2[0];
    alpha[row] = 1.0f / (1.0f + expf(-s));
  }
}

// ---------------------------------------------------------------------------
// Attention: per (b,h) block-column, 8 waves = 8 query rows per block.
// K^T and V staged in LDS; wave-wide shuffle softmax (wave32).
// ---------------------------------------------------------------------------
__global__ __launch_bounds__(256) void attn_kernel(
    const float* __restrict__ qkv, const float* __restrict__ rel_table,
    float* __restrict__ ctx)
{
  __shared__ float sKt[HD * LTOK];   // [d][j]
  __shared__ float sV[LTOK * HD];    // [j][d]
  __shared__ float sP[8][LTOK];
  const int bh = blockIdx.x;
  const int b = bh / NH, h = bh % NH;
  const int tid = threadIdx.x, wave = tid >> 5, lane = tid & 31;

  for (int i = tid; i < LTOK * HD; i += 256) {
    int j = i >> 5, d = i & 31;
    size_t rb = ((size_t)(b * LTOK + j)) * (3 * D) + h * HD + d;
    sKt[d * LTOK + j] = qkv[rb + D];
    sV[i]             = qkv[rb + 2 * D];
  }
  __syncthreads();

  const int r0 = blockIdx.y * 8 + wave;
  const int r  = (r0 < LTOK) ? r0 : (LTOK - 1);
  const size_t grow = (size_t)(b * LTOK + r);
  float q[HD];
#pragma unroll
  for (int d = 0; d < HD; ++d) q[d] = qkv[grow * (3 * D) + h * HD + d];

  const int ri = r / G, ci = r % G;
  float s[7];
  float mx = -3.0e38f;
#pragma unroll
  for (int it = 0; it < 7; ++it) {
    int j = it * 32 + lane;
    float val = -3.0e38f;
    if (j < LTOK) {
      float acc = 0.0f;
#pragma unroll
      for (int d = 0; d < HD; ++d) acc += q[d] * sKt[d * LTOK + j];
      int dr = ri - j / G + (G - 1);
      int dc = ci - j % G + (G - 1);
      val = acc * 0.17677669529663687f + rel_table[(dr * (2 * G - 1) + dc) * NH + h];
    }
    s[it] = val;
    mx = fmaxf(mx, val);
  }
#pragma unroll
  for (int off = 16; off > 0; off >>= 1) mx = fmaxf(mx, __shfl_xor(mx, off, 32));
  float sum = 0.0f;
#pragma unroll
  for (int it = 0; it < 7; ++it) {
    int j = it * 32 + lane;
    if (j < LTOK) {
      float e = expf(s[it] - mx);
      sP[wave][j] = e;
      sum += e;
    }
  }
#pragma unroll
  for (int off = 16; off > 0; off >>= 1) sum += __shfl_xor(sum, off, 32);
  __syncthreads();   // order sP writes before reads (uniform barrier)
  const float inv = 1.0f / sum;
  float acc = 0.0f;
  for (int j = 0; j < LTOK; ++j) acc += sP[wave][j] * sV[j * HD + lane];
  if (r0 < LTOK) ctx[grow * D + h * HD + lane] = acc * inv;
}

// ---------------------------------------------------------------------------
// LayerNorm over D=384, optional residual; one block (384 threads) per row
// ---------------------------------------------------------------------------
__global__ __launch_bounds__(384) void ln_kernel(
    const float* __restrict__ a, const float* __restrict__ res,
    const float* __restrict__ g, const float* __restrict__ beta,
    float* __restrict__ out)
{
  __shared__ float red[D];
  __shared__ float stat[2];
  const int row = blockIdx.x, tid = threadIdx.x;
  const size_t base = (size_t)row * D;
  float v = a[base + tid];
  if (res) v += res[base + tid];
  red[tid] = v;
  __syncthreads();
  for (int s = 192; s >= 3; s >>= 1) { if (tid < s) red[tid] += red[tid + s]; __syncthreads(); }
  if (tid == 0) stat[0] = (red[0] + red[1] + red[2]) * (1.0f / (float)D);
  __syncthreads();
  const float mu = stat[0];
  const float d0 = v - mu;
  red[tid] = d0 * d0;
  __syncthreads();
  for (int s = 192; s >= 3; s >>= 1) { if (tid < s) red[tid] += red[tid + s]; __syncthreads(); }
  if (tid == 0) stat[1] = (red[0] + red[1] + red[2]) * (1.0f / (float)D);
  __syncthreads();
  out[base + tid] = d0 * rsqrtf(stat[1] + 1e-5f) * g[tid] + beta[tid];
}

// ---------------------------------------------------------------------------
// Selective-scan: one block per (direction, batch). Each thread owns 3
// channels x 16 states in registers. All 196 B/C vectors are preloaded into
// LDS once (25 KB of the 320 KB WGP LDS) so the 196-step recurrence runs with
// ZERO barriers; per-step dt/x global loads are software-pipelined one step
// ahead so their latency overlaps the exp/FMA chain.
// dirs: 0 row-fwd, 1 row-rev, 2 col-fwd, 3 col-rev
// ---------------------------------------------------------------------------
__global__ __launch_bounds__(256) void scan_kernel(
    const float* __restrict__ xz, const float* __restrict__ x_dbl,
    const float* __restrict__ dt, const float* __restrict__ A_log,
    float* __restrict__ y4)
{
  const int blk = blockIdx.x;      // 0..31
  const int b = blk & 7;
  const int dir = blk >> 3;
  const int tid = threadIdx.x;
  __shared__ float sB[LTOK][NS];
  __shared__ float sC[LTOK][NS];

  // preload B/C for the whole sequence (token-indexed; order-independent)
  for (int i = tid; i < LTOK * NS; i += 256) {
    int tok = i >> 4, n = i & 15;
    size_t row = (size_t)(b * LTOK + tok);
    sB[tok][n] = x_dbl[row * 56 + DT_RANK + n];
    sC[tok][n] = x_dbl[row * 56 + DT_RANK + NS + n];
  }

  float Aneg[3][NS];
  float h[3][NS];
#pragma unroll
  for (int i = 0; i < 3; ++i) {
    int e = tid + 256 * i;
#pragma unroll
    for (int n = 0; n < NS; ++n) {
      Aneg[i][n] = -expf(A_log[(size_t)e * NS + n]);
      h[i][n] = 0.0f;
    }
  }
  __syncthreads();   // single barrier: B/C resident for the whole scan

  auto tok_of = [&](int t) -> int {
    int p = (dir == 1 || dir == 3) ? (LTOK - 1 - t) : t;
    return (dir >= 2) ? ((p % G) * G + p / G) : p;
  };

  int tok = tok_of(0);
  size_t row = (size_t)(b * LTOK + tok);
  float dtv[3], xv[3];
#pragma unroll
  for (int i = 0; i < 3; ++i) {
    int e = tid + 256 * i;
    dtv[i] = dt[row * E + e];
    xv[i]  = xz[row * (2 * E) + e];
  }

  for (int t = 0; t < LTOK; ++t) {
    const int tokn = tok_of((t + 1 < LTOK) ? (t + 1) : t);
    const size_t rown = (size_t)(b * LTOK + tokn);
    float dtn[3], xn[3];
#pragma unroll
    for (int i = 0; i < 3; ++i) {       // issue next step's loads early
      int e = tid + 256 * i;
      dtn[i] = dt[rown * E + e];
      xn[i]  = xz[rown * (2 * E) + e];
    }
#pragma unroll
    for (int i = 0; i < 3; ++i) {
      int e = tid + 256 * i;
      float acc = 0.0f;
#pragma unroll
      for (int n = 0; n < NS; ++n) {
        float ab = expf(dtv[i] * Aneg[i][n]);
        h[i][n] = ab * h[i][n] + dtv[i] * sB[tok][n] * xv[i];
        acc += h[i][n] * sC[tok][n];
      }
      y4[((size_t)dir * MROWS + row) * E + e] = acc;
    }
    tok = tokn; row = rown;
#pragma unroll
    for (int i = 0; i < 3; ++i) { dtv[i] = dtn[i]; xv[i] = xn[i]; }
  }
}

// ---------------------------------------------------------------------------
// Elementwise kernels
// ---------------------------------------------------------------------------
__global__ __launch_bounds__(256) void gate_kernel(
    const float* __restrict__ y4, const float* __restrict__ xz,
    const float* __restrict__ dskip, float* __restrict__ ygate)
{
  const size_t i = (size_t)blockIdx.x * 256 + threadIdx.x;
  const size_t tot = (size_t)MROWS * E;
  if (i < tot) {
    const size_t row = i / E;
    const int e = (int)(i % E);
    float y = 0.25f * (y4[i] + y4[tot + i] + y4[2 * tot + i] + y4[3 * tot + i]);
    float z   = xz[row * (2 * E) + E + e];
    float xin = xz[row * (2 * E) + e];
    float sil = z / (1.0f + expf(-z));
    ygate[i] = y * sil + xin * dskip[e];
  }
}

__global__ __launch_bounds__(256) void mix_kernel(
    const float* __restrict__ alpha, const float* __restrict__ attn_out,
    const float* __restrict__ vssm_out, float* __restrict__ ymix)
{
  const size_t i = (size_t)blockIdx.x * 256 + threadIdx.x;
  if (i < (size_t)MROWS * D) {
    const float a = alpha[i / D];
    ymix[i] = a * attn_out[i] + (1.0f - a) * vssm_out[i];
  }
}

__global__ __launch_bounds__(256) void add_kernel(
    const float* __restrict__ a, const float* __restrict__ b, float* __restrict__ o)
{
  const size_t i = (size_t)blockIdx.x * 256 + threadIdx.x;
  if (i < (size_t)MROWS * D) o[i] = a[i] + b[i];
}

// ---------------------------------------------------------------------------
// Host-side launch
// ---------------------------------------------------------------------------
extern "C" void kernel_launch(void* const* d_in, const int* in_sizes, int n_in,
                              void* d_out, int out_size, void* d_ws, size_t ws_size,
                              hipStream_t stream) {
  (void)in_sizes; (void)n_in; (void)out_size; (void)ws_size;
  const float* x        = (const float*)d_in[0];
  const float* entropy  = (const float*)d_in[1];
  const float* rw1      = (const float*)d_in[2];
  const float* rb1      = (const float*)d_in[3];
  const float* rw2      = (const float*)d_in[4];
  const float* rb2      = (const float*)d_in[5];
  const float* qkv_w    = (const float*)d_in[6];
  const float* qkv_b    = (const float*)d_in[7];
  const float* aproj_w  = (const float*)d_in[8];
  const float* aproj_b  = (const float*)d_in[9];
  const float* ang      = (const float*)d_in[10];
  const float* anb      = (const float*)d_in[11];
  const float* rel      = (const float*)d_in[12];
  const float* inproj_w = (const float*)d_in[13];
  const float* A_log    = (const float*)d_in[14];
  const float* xproj_w  = (const float*)d_in[15];
  const float* dtw      = (const float*)d_in[16];
  const float* dtb      = (const float*)d_in[17];
  const float* dskip    = (const float*)d_in[18];
  const float* oproj_w  = (const float*)d_in[19];
  const float* vng      = (const float*)d_in[20];
  const float* vnb      = (const float*)d_in[21];
  const float* fng      = (const float*)d_in[22];
  const float* fnb      = (const float*)d_in[23];
  const float* fw1      = (const float*)d_in[24];
  const float* fb1      = (const float*)d_in[25];
  const float* fw2      = (const float*)d_in[26];
  const float* fb2      = (const float*)d_in[27];
  float* out = (float*)d_out;
  float* ws  = (float*)d_ws;

  size_t o = 0;
  float* qkv      = ws + o; o += (size_t)MROWS * 3 * D;     // 1568x1152
  float* ctx      = ws + o; o += (size_t)MROWS * D;
  float* aproj    = ws + o; o += (size_t)MROWS * D;
  float* attn_out = ws + o; o += (size_t)MROWS * D;
  float* xz       = ws + o; o += (size_t)MROWS * 2 * E;     // 1568x1536
  float* xdbl     = ws + o; o += (size_t)MROWS * 56;
  float* dt       = ws + o; o += (size_t)MROWS * E;
  float* y4       = ws + o; o += (size_t)4 * MROWS * E;
  float* ygate    = ws + o; o += (size_t)MROWS * E;
  float* vproj    = ws + o; o += (size_t)MROWS * D;
  float* vout     = ws + o; o += (size_t)MROWS * D;
  float* alpha    = ws + o; o += (size_t)MROWS;
  float* ymix     = ws + o; o += (size_t)MROWS * D;
  float* hnorm    = ws + o; o += (size_t)MROWS * D;
  float* ffnh     = ws + o; o += (size_t)MROWS * FFN_DIM;
  float* ffn2     = ws + o; o += (size_t)MROWS * D;

  auto gemm = [&](const float* A, int lda, const float* W, int ldb, float* C, int ldc,
                  const float* bias, int N, int K, int epi) {
    dim3 grid((N + BN - 1) / BN, (MROWS + BM - 1) / BM);
    gemm_wmma_f32<<<grid, 128, 0, stream>>>(A, lda, W, ldb, C, ldc, bias, MROWS, N, K, epi);
  };

  // router
  router_kernel<<<MROWS, 128, 0, stream>>>(x, entropy, rw1, rb1, rw2, rb2, alpha);
  // attention branch
  gemm(x, D, qkv_w, D, qkv, 3 * D, qkv_b, 3 * D, D, 1);
  attn_kernel<<<dim3(B_SZ * NH, (LTOK + 7) / 8), 256, 0, stream>>>(qkv, rel, ctx);
  gemm(ctx, D, aproj_w, D, aproj, D, aproj_b, D, D, 1);
  ln_kernel<<<MROWS, D, 0, stream>>>(aproj, x, ang, anb, attn_out);
  // vssm branch
  gemm(x, D, inproj_w, D, xz, 2 * E, nullptr, 2 * E, D, 0);
  gemm(xz, 2 * E, xproj_w, E, xdbl, 56, nullptr, 56, E, 0);
  gemm(xdbl, 56, dtw, DT_RANK, dt, E, dtb, E, DT_RANK, 3);
  scan_kernel<<<32, 256, 0, stream>>>(xz, xdbl, dt, A_log, y4);
  gate_kernel<<<((size_t)MROWS * E + 255) / 256, 256, 0, stream>>>(y4, xz, dskip, ygate);
  gemm(ygate, E, oproj_w, E, vproj, D, nullptr, D, E, 0);
  ln_kernel<<<MROWS, D, 0, stream>>>(vproj, x, vng, vnb, vout);
  // mix + FFN
  mix_kernel<<<((size_t)MROWS * D + 255) / 256, 256, 0, stream>>>(alpha, attn_out, vout, ymix);
  ln_kernel<<<MROWS, D, 0, stream>>>(ymix, nullptr, fng, fnb, hnorm);
  gemm(hnorm, D, fw1, D, ffnh, FFN_DIM, fb1, FFN_DIM, D, 2);
  gemm(ffnh, FFN_DIM, fw2, FFN_DIM, ffn2, D, fb2, D, FFN_DIM, 1);
  add_kernel<<<((size_t)MROWS * D + 255) / 256, 256, 0, stream>>>(ymix, ffn2, out);
}